// Sae_44590350467557
// MI455X (gfx1250) — compile-verified
//
#include <hip/hip_runtime.h>
#include <hip/hip_bf16.h>
#include <stddef.h>

#define N_TOK 8192
#define D_IN  2048
#define D_SAE 16384
#define KTOP  32

typedef float v2f __attribute__((ext_vector_type(2)));
typedef float v8f __attribute__((ext_vector_type(8)));

// ---- CDNA5 async global->LDS path (signature learned from clang diagnostic:
//      (v4i __device__* src, v4i as3* lds, imm offset, imm cpol)) ----
typedef int v4i __attribute__((vector_size(16)));
typedef __attribute__((address_space(1))) v4i g_v4i;
typedef __attribute__((address_space(3))) v4i l_v4i;

#if __has_builtin(__builtin_amdgcn_global_load_async_to_lds_b128)
#define HAVE_ASYNC_LDS 1
#define ASYNC_LOAD_B128(g, l) \
    __builtin_amdgcn_global_load_async_to_lds_b128((g_v4i*)(g), (l_v4i*)(l), 0, 0)
#else
#define HAVE_ASYNC_LDS 0
#endif

#if __has_builtin(__builtin_amdgcn_s_wait_asynccnt)
#define WAIT_ASYNC0() __builtin_amdgcn_s_wait_asynccnt(0)
#else
#define WAIT_ASYNC0() asm volatile("s_wait_asynccnt 0x0" ::: "memory")
#endif

// ---------------------------------------------------------------------------
// Kernel 0a/0b: bias[n] = b_enc[n] - sum_k b_dec[k] * W_enc[k,n]
// (folds the "- b_dec" input shift into the encode epilogue so the GEMM can
//  stream raw x tiles straight into LDS)
// ---------------------------------------------------------------------------
#define BIAS_SPLIT 16

__global__ __launch_bounds__(256)
void sae_bias_partial(const float* __restrict__ b_dec,
                      const float* __restrict__ W_enc,
                      float* __restrict__ partial) {
    const int n  = blockIdx.x * 256 + threadIdx.x;
    const int k0 = blockIdx.y * (D_IN / BIAS_SPLIT);
    float acc = 0.0f;
    for (int k = k0; k < k0 + D_IN / BIAS_SPLIT; ++k)
        acc += b_dec[k] * W_enc[(size_t)k * D_SAE + n];
    partial[(size_t)blockIdx.y * D_SAE + n] = acc;
}

__global__ __launch_bounds__(256)
void sae_bias_reduce(const float* __restrict__ partial,
                     const float* __restrict__ b_enc,
                     float* __restrict__ bias) {
    const int n = blockIdx.x * 256 + threadIdx.x;
    float acc = 0.0f;
    #pragma unroll
    for (int j = 0; j < BIAS_SPLIT; ++j)
        acc += partial[(size_t)j * D_SAE + n];
    bias[n] = b_enc[n] - acc;
}

// ---------------------------------------------------------------------------
// Kernel 1: feats = relu(x @ W_enc + bias)   [N_TOK, D_SAE]
// WMMA f32 16x16x4. Block tile 64(M) x 128(N), 8 waves (2x4), each wave owns
// a 32x32 quad of C tiles. K chunks of 32 streamed global->LDS with
// GLOBAL_LOAD_ASYNC_TO_LDS_B128 (double-buffered, zero staging VGPRs).
// ---------------------------------------------------------------------------
#define BM 64
#define BN 128
#define BK 32
#define LDA 36    // BK + 4 pad (144B row stride, 16B aligned)
#define LDB 132   // BN + 4 pad (528B row stride, 16B aligned)
#define NCHUNK (D_IN / BK)   // 64 (even)

__global__ __launch_bounds__(256)
void sae_encode_gemm(const float* __restrict__ x,
                     const float* __restrict__ W_enc,
                     const float* __restrict__ bias,
                     float* __restrict__ feats) {
    __shared__ float AsBuf[2 * BM * LDA];
    __shared__ float BsBuf[2 * BK * LDB];

    const int t    = threadIdx.x;
    const int lane = t & 31;
    const int wave = t >> 5;
    const int mrow = wave >> 2;     // 0..1 -> M offset mrow*32
    const int ncol = wave & 3;      // 0..3 -> N offset ncol*32
    const int half = lane >> 4;     // K-pair select (K=0,1 vs K=2,3)
    const int l15  = lane & 15;

    const int n0 = blockIdx.x * BN;
    const int m0 = blockIdx.y * BM;

    v8f acc[2][2] = {};

    auto compute_chunk = [&](int buf) {
        const float* As = AsBuf + buf * (BM * LDA);
        const float* Bs = BsBuf + buf * (BK * LDB);
        #pragma unroll
        for (int kk = 0; kk < BK / 4; ++kk) {
            const int kb = kk * 4 + half * 2;    // this lane-half holds K=kb, kb+1
            v2f a[2], b[2];
            #pragma unroll
            for (int mi = 0; mi < 2; ++mi) {
                const int m = mrow * 32 + mi * 16 + l15;
                a[mi].x = As[m * LDA + kb];
                a[mi].y = As[m * LDA + kb + 1];
            }
            #pragma unroll
            for (int ni = 0; ni < 2; ++ni) {
                const int n = ncol * 32 + ni * 16 + l15;
                b[ni].x = Bs[kb * LDB + n];
                b[ni].y = Bs[(kb + 1) * LDB + n];
            }
            acc[0][0] = __builtin_amdgcn_wmma_f32_16x16x4_f32(false, a[0], false, b[0],
                                                              (short)0, acc[0][0], false, false);
            acc[0][1] = __builtin_amdgcn_wmma_f32_16x16x4_f32(false, a[0], false, b[1],
                                                              (short)0, acc[0][1], false, false);
            acc[1][0] = __builtin_amdgcn_wmma_f32_16x16x4_f32(false, a[1], false, b[0],
                                                              (short)0, acc[1][0], false, false);
            acc[1][1] = __builtin_amdgcn_wmma_f32_16x16x4_f32(false, a[1], false, b[1],
                                                              (short)0, acc[1][1], false, false);
        }
    };

#if HAVE_ASYNC_LDS
    // per-thread fixed copy lanes (A: 2x b128, B: 4x b128 per chunk)
    auto async_load_chunk = [&](int buf, int k0) {
        float* As = AsBuf + buf * (BM * LDA);
        float* Bs = BsBuf + buf * (BK * LDB);
        #pragma unroll
        for (int j = 0; j < 2; ++j) {            // A: 64x32 = 512 float4
            const int i   = t + j * 256;
            const int row = i >> 3;
            const int c4  = i & 7;
            ASYNC_LOAD_B128(x + (size_t)(m0 + row) * D_IN + k0 + c4 * 4,
                            As + row * LDA + c4 * 4);
        }
        #pragma unroll
        for (int j = 0; j < 4; ++j) {            // B: 32x128 = 1024 float4
            const int i  = t + j * 256;
            const int kr = i >> 5;
            const int c4 = i & 31;
            ASYNC_LOAD_B128(W_enc + (size_t)(k0 + kr) * D_SAE + n0 + c4 * 4,
                            Bs + kr * LDB + c4 * 4);
        }
    };

    async_load_chunk(0, 0);
    WAIT_ASYNC0();
    __syncthreads();
    for (int c = 0; c < NCHUNK; c += 2) {
        async_load_chunk(1, (c + 1) * BK);       // next chunk streams during compute
        compute_chunk(0);
        WAIT_ASYNC0();
        __syncthreads();
        if (c + 2 < NCHUNK) async_load_chunk(0, (c + 2) * BK);
        compute_chunk(1);
        WAIT_ASYNC0();
        __syncthreads();
    }
#else
    // fallback: synchronous single-buffer staging (no long-lived staging regs)
    for (int k0 = 0; k0 < D_IN; k0 += BK) {
        #pragma unroll
        for (int j = 0; j < 2; ++j) {
            const int i   = t + j * 256;
            const int row = i >> 3;
            const int c4  = i & 7;
            *(float4*)(&AsBuf[row * LDA + c4 * 4]) =
                *(const float4*)(x + (size_t)(m0 + row) * D_IN + k0 + c4 * 4);
        }
        #pragma unroll
        for (int j = 0; j < 4; ++j) {
            const int i  = t + j * 256;
            const int kr = i >> 5;
            const int c4 = i & 31;
            *(float4*)(&BsBuf[kr * LDB + c4 * 4]) =
                *(const float4*)(W_enc + (size_t)(k0 + kr) * D_SAE + n0 + c4 * 4);
        }
        __syncthreads();
        compute_chunk(0);
        __syncthreads();
    }
#endif

    // ---- epilogue: +bias, relu, store dense feats ----
    #pragma unroll
    for (int mi = 0; mi < 2; ++mi) {
        #pragma unroll
        for (int ni = 0; ni < 2; ++ni) {
            const int gn = n0 + ncol * 32 + ni * 16 + l15;
            const float be = bias[gn];
            #pragma unroll
            for (int r = 0; r < 8; ++r) {
                const int gm = m0 + mrow * 32 + mi * 16 + r + half * 8;
                const float v = acc[mi][ni][r] + be;
                feats[(size_t)gm * D_SAE + gn] = v > 0.0f ? v : 0.0f;
            }
        }
    }
}

// ---------------------------------------------------------------------------
// Kernel 2: per-row top-32 of 16384 (feats >= 0 post-relu, so -1 = sentinel).
// Each thread holds 64 candidates in registers; wave32 shfl argmax reduce,
// cross-wave via tiny LDS. Row rewritten masked in place; (val,idx) -> ws.
// ---------------------------------------------------------------------------
__global__ __launch_bounds__(256)
void sae_topk(float* __restrict__ feats,
              float* __restrict__ top_vals,
              int* __restrict__ top_idx) {
    __shared__ float rv[8];
    __shared__ int   ri[8];
    __shared__ float topv[KTOP];
    __shared__ int   topi[KTOP];
    __shared__ int   wbi_s;

    const int t    = threadIdx.x;
    const int lane = t & 31;
    const int wave = t >> 5;
    const size_t row = blockIdx.x;
    float* frow = feats + row * (size_t)D_SAE;

    float loc[64];
    #pragma unroll
    for (int s = 0; s < 64; ++s) loc[s] = frow[t + (s << 8)];

    for (int it = 0; it < KTOP; ++it) {
        float bv = -1.0f;
        int   bi = t;
        #pragma unroll
        for (int s = 0; s < 64; ++s) {
            const float v = loc[s];
            if (v > bv) { bv = v; bi = t + (s << 8); }
        }
        #pragma unroll
        for (int off = 16; off > 0; off >>= 1) {
            const float ov = __shfl_down(bv, off, 32);
            const int   oi = __shfl_down(bi, off, 32);
            if (ov > bv) { bv = ov; bi = oi; }
        }
        if (lane == 0) { rv[wave] = bv; ri[wave] = bi; }
        __syncthreads();
        if (t == 0) {
            float mv = rv[0]; int mi = ri[0];
            #pragma unroll
            for (int w = 1; w < 8; ++w)
                if (rv[w] > mv) { mv = rv[w]; mi = ri[w]; }
            topv[it] = mv; topi[it] = mi; wbi_s = mi;
        }
        __syncthreads();
        const int wbi = wbi_s;
        if ((wbi & 255) == t) {
            const int slot = wbi >> 8;
            #pragma unroll
            for (int s = 0; s < 64; ++s)
                if (s == slot) loc[s] = -1.0f;
        }
        __syncthreads();
    }

    // masked rewrite: zero everything, then scatter the 32 winners
    #pragma unroll
    for (int s = 0; s < 64; ++s) frow[t + (s << 8)] = 0.0f;
    __threadfence_block();
    __syncthreads();
    if (t < KTOP) {
        frow[topi[t]] = topv[t];
        top_vals[row * KTOP + t] = topv[t];
        top_idx[row * KTOP + t]  = topi[t];
    }
}

// ---------------------------------------------------------------------------
// Kernel 3: sae_out[n,:] = b_dec + sum_j top_val_j * W_dec[top_idx_j, :]
// ---------------------------------------------------------------------------
__global__ __launch_bounds__(256)
void sae_decode(const float* __restrict__ top_vals,
                const int* __restrict__ top_idx,
                const float* __restrict__ W_dec,
                const float* __restrict__ b_dec,
                float* __restrict__ sae_out) {
    __shared__ float sv[KTOP];
    __shared__ int   si[KTOP];
    const int t = threadIdx.x;
    const size_t n = blockIdx.x;
    if (t < KTOP) {
        sv[t] = top_vals[n * KTOP + t];
        si[t] = top_idx[n * KTOP + t];
    }
    __syncthreads();
    #pragma unroll
    for (int h = 0; h < 2; ++h) {
        const int c4 = t + h * 256;            // 512 float4 per row
        float4 acc = ((const float4*)b_dec)[c4];
        #pragma unroll
        for (int j = 0; j < KTOP; ++j) {
            const float  v = sv[j];
            const float4 w = ((const float4*)(W_dec + (size_t)si[j] * D_IN))[c4];
            acc.x += v * w.x; acc.y += v * w.y;
            acc.z += v * w.z; acc.w += v * w.w;
        }
        ((float4*)(sae_out + n * D_IN))[c4] = acc;
    }
}

// ---------------------------------------------------------------------------
// Kernel 4: per-column err = sum_n (out-x)^2, sum = sum_n x, sq = sum_n x^2
// ---------------------------------------------------------------------------
__global__ __launch_bounds__(256)
void sae_colstats(const float* __restrict__ x,
                  const float* __restrict__ sae_out,
                  float* __restrict__ col_err,
                  float* __restrict__ col_sum,
                  float* __restrict__ col_sq) {
    const int c = blockIdx.x * 256 + threadIdx.x;
    float err = 0.0f, s = 0.0f, sq = 0.0f;
    for (int n = 0; n < N_TOK; ++n) {
        const float xv = x[(size_t)n * D_IN + c];
        const float ov = sae_out[(size_t)n * D_IN + c];
        const float d  = ov - xv;
        err += d * d;
        s   += xv;
        sq  += xv * xv;
    }
    col_err[c] = err;
    col_sum[c] = s;
    col_sq[c]  = sq;
}

// ---------------------------------------------------------------------------
// Kernel 5: fvu = mean_c( err[c] / (sq[c] - sum[c]^2 / N) )
// ---------------------------------------------------------------------------
__global__ __launch_bounds__(256)
void sae_fvu(const float* __restrict__ col_err,
             const float* __restrict__ col_sum,
             const float* __restrict__ col_sq,
             float* __restrict__ out) {
    __shared__ float red[256];
    const int t = threadIdx.x;
    float acc = 0.0f;
    for (int c = t; c < D_IN; c += 256) {
        const float tv = col_sq[c] - col_sum[c] * col_sum[c] * (1.0f / N_TOK);
        acc += col_err[c] / tv;
    }
    red[t] = acc;
    __syncthreads();
    for (int s = 128; s > 0; s >>= 1) {
        if (t < s) red[t] += red[t + s];
        __syncthreads();
    }
    if (t == 0) out[0] = red[0] * (1.0f / D_IN);
}

// ---------------------------------------------------------------------------
extern "C" void kernel_launch(void* const* d_in, const int* in_sizes, int n_in,
                              void* d_out, int out_size, void* d_ws, size_t ws_size,
                              hipStream_t stream) {
    (void)in_sizes; (void)n_in; (void)out_size; (void)ws_size;

    const float* x     = (const float*)d_in[0];
    const float* W_enc = (const float*)d_in[1];
    const float* b_enc = (const float*)d_in[2];
    const float* W_dec = (const float*)d_in[3];
    const float* b_dec = (const float*)d_in[4];
    // d_in[5] = k (always 32, compile-time constant here)

    float* out      = (float*)d_out;
    float* sae_out  = out;                                           // [N_TOK * D_IN]
    float* feats    = out + (size_t)N_TOK * D_IN;                    // [N_TOK * D_SAE]
    float* fvu      = out + (size_t)N_TOK * D_IN + (size_t)N_TOK * D_SAE;  // [1]

    float* wsf      = (float*)d_ws;
    float* top_vals = wsf;                                           // [N_TOK * KTOP]
    int*   top_idx  = (int*)(wsf + (size_t)N_TOK * KTOP);            // [N_TOK * KTOP]
    float* col_err  = wsf + (size_t)2 * N_TOK * KTOP;                // [D_IN]
    float* col_sum  = col_err + D_IN;                                // [D_IN]
    float* col_sq   = col_sum + D_IN;                                // [D_IN]
    float* bias     = col_sq + D_IN;                                 // [D_SAE]
    float* partial  = bias + D_SAE;                                  // [BIAS_SPLIT * D_SAE]

    dim3 bias_grid(D_SAE / 256, BIAS_SPLIT);
    sae_bias_partial<<<bias_grid, 256, 0, stream>>>(b_dec, W_enc, partial);
    sae_bias_reduce<<<D_SAE / 256, 256, 0, stream>>>(partial, b_enc, bias);

    dim3 gemm_grid(D_SAE / BN, N_TOK / BM);   // 128 x 128 blocks
    sae_encode_gemm<<<gemm_grid, 256, 0, stream>>>(x, W_enc, bias, feats);
    sae_topk<<<N_TOK, 256, 0, stream>>>(feats, top_vals, top_idx);
    sae_decode<<<N_TOK, 256, 0, stream>>>(top_vals, top_idx, W_dec, b_dec, sae_out);
    sae_colstats<<<D_IN / 256, 256, 0, stream>>>(x, sae_out, col_err, col_sum, col_sq);
    sae_fvu<<<1, 256, 0, stream>>>(col_err, col_sum, col_sq, fvu);
}